// CapsuleLayer_21071109554325
// MI455X (gfx1250) — compile-verified
//
#include <hip/hip_runtime.h>
#include <math.h>

// Problem constants (from reference): x[64,2048,16], W[2048,32,16,16], 3 routing iters.
#define BATCH 64
#define INC   2048
#define NJ    32
#define DIM   16
#define ICH   16                 // input capsules per block
#define NBLK  (INC / ICH)        // 128 blocks per pass
#define SZ    (BATCH * NJ * DIM) // 32768 = size of s / outsum / output
#define NTHR  512                // 16 waves: 4 m-tiles x 4 j-groups
#define JT    8                  // j-tiles per wave

typedef float v2f __attribute__((ext_vector_type(2)));
typedef float v8f __attribute__((ext_vector_type(8)));

// Sum across the 16-lane (d) group using DPP row_xmask butterflies (pure VALU,
// no LDS/dscnt stalls). DPP_ROW_XMASK = 0x160 | mask.
#define DPP_XOR(x, m)                                                          \
  __int_as_float(__builtin_amdgcn_update_dpp(                                  \
      0, __float_as_int(x), 0x160 | (m), 0xf, 0xf, true))

__device__ __forceinline__ float xorsum16(float p) {
  p += DPP_XOR(p, 1);
  p += DPP_XOR(p, 2);
  p += DPP_XOR(p, 4);
  p += DPP_XOR(p, 8);
  return p;  // all 16 lanes of the group hold the sum
}

__global__ void caps_init(float* p, int n) {
  int i = blockIdx.x * blockDim.x + threadIdx.x;
  if (i < n) p[i] = 0.f;
}

// One routing pass, fused. Phase 1: u_hat tiles via V_WMMA_F32_16X16X4_F32,
// logits = u_hat . outsum (outsum==0 -> uniform softmax, matching iter 0).
// Phase 2: softmax coupling folded into the A operand (A' = c o X) and the
// weighted sum s += c*u_hat computed as a second WMMA accumulating into sacc.
// MODEA: per-block partials to ws. else atomics.
template <bool MODEA>
__global__ __launch_bounds__(NTHR) void caps_pass(const float* __restrict__ x,
                                                  const float* __restrict__ W,
                                                  const float* __restrict__ outsum,
                                                  float* sdst) {
  __shared__ float Xs[BATCH * DIM];     // x[:, i, :]            4 KB
  __shared__ float Ws[NJ * DIM * DIM];  // W[i, :, :, :]        32 KB
  __shared__ float Lg[BATCH * NJ];      // logits for current i  8 KB
  __shared__ float smax[BATCH];
  __shared__ float sinv[BATCH];

  const int tid  = threadIdx.x;
  const int lane = tid & 31;
  const int wv   = tid >> 5;        // 16 waves
  const int half = lane >> 4;       // 0: lanes 0-15, 1: lanes 16-31
  const int l16  = lane & 15;       // N-index (d) in B/C/D frags; M-row in A
  const int mt   = wv & 3;          // batch m-tile (16 batches each)
  const int jb   = (wv >> 2) * JT;  // j range [jb, jb+8)
  const int i0   = blockIdx.x * ICH;

  const v8f vzero = {0.f, 0.f, 0.f, 0.f, 0.f, 0.f, 0.f, 0.f};
  v8f sacc[JT];  // WMMA C/D accumulators for s contributions
#pragma unroll
  for (int jj = 0; jj < JT; ++jj) sacc[jj] = vzero;

  // Preload outsum operands: invariant across the whole i-loop.
  // osum[jj][v] pairs with D-frag element v (b = mt*16+v+8*half, d = l16).
  float osum[JT][8];
#pragma unroll
  for (int jj = 0; jj < JT; ++jj)
#pragma unroll
    for (int v = 0; v < 8; ++v) {
      const int b = mt * 16 + v + 8 * half;
      osum[jj][v] = outsum[((size_t)b * NJ + jb + jj) * DIM + l16];
    }

  for (int ii = 0; ii < ICH; ++ii) {
    const int i = i0 + ii;
    // stage x[:, i, :] : 64 rows x 16 floats, 8 threads per row (float2 each)
    {
      const int r = tid >> 3, c = (tid & 7) * 2;
      const float2 vx = *(const float2*)(x + ((size_t)r * INC + i) * DIM + c);
      *(float2*)(Xs + r * DIM + c) = vx;
    }
    // stage W[i] : contiguous 32 KB, fully coalesced float4 copy
    {
      const float4* src = (const float4*)(W + (size_t)i * NJ * DIM * DIM);
      float4* dst = (float4*)Ws;
#pragma unroll
      for (int q = 0; q < 4; ++q) dst[q * NTHR + tid] = src[q * NTHR + tid];
    }
    if (i + 1 < INC)  // pull next W slab toward L2 (global_prefetch_b8)
      __builtin_prefetch(W + (size_t)(i + 1) * NJ * DIM * DIM, 0, 1);
    __syncthreads();

    // A fragments: A = X tile (M=16 batches of this m-tile, K=16).
    // 16x4 f32 A layout: lanes0-15 M=0..15 {K=4kk,4kk+1}; lanes16-31 {K=4kk+2,4kk+3}
    v2f afr[4];
#pragma unroll
    for (int kk = 0; kk < 4; ++kk) {
      const int base = (mt * 16 + l16) * DIM + 4 * kk + 2 * half;
      afr[kk].x = Xs[base];
      afr[kk].y = Xs[base + 1];
    }

    // ---- Phase 1: u_hat tiles -> logits (u_hat transient) ----
#pragma unroll
    for (int jj = 0; jj < JT; ++jj) {
      const int j = jb + jj;
      v8f acc = vzero;
#pragma unroll
      for (int kk = 0; kk < 4; ++kk) {
        // B = W[i,j]^T (K=k rows, N=d cols): VGPR0 rows {4kk,4kk+2}, VGPR1 {+1,+3}
        v2f bfr;
        const int base = (j * DIM + l16) * DIM + 4 * kk + 2 * half;
        bfr.x = Ws[base];
        bfr.y = Ws[base + 1];
        acc = __builtin_amdgcn_wmma_f32_16x16x4_f32(false, afr[kk], false, bfr,
                                                    (short)0, acc, false, false);
      }
      // logit[b,i,j] = sum_d u_hat * outsum[b,j,d]; VALU butterfly over d group
#pragma unroll
      for (int v = 0; v < 8; ++v) {
        const float p = xorsum16(acc[v] * osum[jj][v]);
        if (l16 == 0) Lg[(mt * 16 + v + 8 * half) * NJ + j] = p;
      }
    }
    __syncthreads();

    // softmax stats over j (one thread per batch)
    if (tid < BATCH) {
      float m = -3.4e38f;
      for (int j = 0; j < NJ; ++j) m = fmaxf(m, Lg[tid * NJ + j]);
      float s = 0.f;
      for (int j = 0; j < NJ; ++j) s += expf(Lg[tid * NJ + j] - m);
      smax[tid] = m;
      sinv[tid] = 1.f / s;
    }
    __syncthreads();

    // ---- Phase 2: s += c * u_hat as WMMA with row-scaled A' = c o X ----
    const int brow = mt * 16 + l16;  // A-matrix row owned by this lane
    const float mrow = smax[brow];
    const float srow = sinv[brow];
#pragma unroll
    for (int jj = 0; jj < JT; ++jj) {
      const int j = jb + jj;
      const float c = expf(Lg[brow * NJ + j] - mrow) * srow;  // c[brow, i, j]
#pragma unroll
      for (int kk = 0; kk < 4; ++kk) {
        v2f afs, bfr;
        afs.x = c * afr[kk].x;
        afs.y = c * afr[kk].y;
        const int base = (j * DIM + l16) * DIM + 4 * kk + 2 * half;
        bfr.x = Ws[base];
        bfr.y = Ws[base + 1];
        sacc[jj] = __builtin_amdgcn_wmma_f32_16x16x4_f32(
            false, afs, false, bfr, (short)0, sacc[jj], false, false);
      }
    }
    __syncthreads();  // before next iteration overwrites Xs/Ws/Lg
  }

  if (MODEA) {
    float* dst = sdst + (size_t)blockIdx.x * SZ;
#pragma unroll
    for (int jj = 0; jj < JT; ++jj)
#pragma unroll
      for (int v = 0; v < 8; ++v) {
        const int b = mt * 16 + v + 8 * half;
        dst[((size_t)b * NJ + jb + jj) * DIM + l16] = sacc[jj][v];
      }
  } else {
#pragma unroll
    for (int jj = 0; jj < JT; ++jj)
#pragma unroll
      for (int v = 0; v < 8; ++v) {
        const int b = mt * 16 + v + 8 * half;
        unsafeAtomicAdd(&sdst[((size_t)b * NJ + jb + jj) * DIM + l16], sacc[jj][v]);
      }
  }
}

// Reduce partials (if any), squash over d (16-lane group), accumulate outsum,
// optionally write final output; zero the atomic s buffer for the next pass.
__global__ void caps_squash(const float* src, float* outsum, float* dout,
                            float* szero, int nparts) {
  const int idx = blockIdx.x * blockDim.x + threadIdx.x;  // < SZ, d = idx & 15
  float v = 0.f;
  for (int p = 0; p < nparts; ++p) v += src[(size_t)p * SZ + idx];
  if (szero) szero[idx] = 0.f;
  const float sq = xorsum16(v * v);
  const float coef = sq / ((1.f + sq) * sqrtf(sq + 1e-7f));
  const float o = coef * v;
  outsum[idx] += o;
  if (dout) dout[idx] = o;
}

extern "C" void kernel_launch(void* const* d_in, const int* in_sizes, int n_in,
                              void* d_out, int out_size, void* d_ws, size_t ws_size,
                              hipStream_t stream) {
  const float* x = (const float*)d_in[0];
  const float* W = (const float*)d_in[1];
  float* out = (float*)d_out;
  float* ws = (float*)d_ws;

  float* outsum = ws;            // SZ floats
  float* sbuf   = ws + SZ;       // SZ floats (atomic fallback)
  float* parts  = ws + 2 * SZ;   // NBLK*SZ floats (preferred, atomic-free)

  const bool modeA = ws_size >= (size_t)(2 * SZ + NBLK * SZ) * sizeof(float);

  int ninit = 2 * SZ;
  if ((size_t)ninit * sizeof(float) > ws_size) ninit = (int)(ws_size / sizeof(float));
  caps_init<<<(ninit + 255) / 256, 256, 0, stream>>>(ws, ninit);

  for (int t = 0; t < 3; ++t) {
    if (modeA)
      caps_pass<true><<<NBLK, NTHR, 0, stream>>>(x, W, outsum, parts);
    else
      caps_pass<false><<<NBLK, NTHR, 0, stream>>>(x, W, outsum, sbuf);
    caps_squash<<<SZ / 256, 256, 0, stream>>>(modeA ? parts : sbuf, outsum,
                                              (t == 2) ? out : nullptr,
                                              modeA ? nullptr : sbuf,
                                              modeA ? NBLK : 1);
  }
}